// recon_encoder_26680336843514
// MI455X (gfx1250) — compile-verified
//
#include <hip/hip_runtime.h>

typedef __attribute__((ext_vector_type(2))) float v2f;
typedef __attribute__((ext_vector_type(8))) float v8f;

// ---------------------------------------------------------------------------
// Edge scatter: summed[dst] += h[src]  (and cnt[dst] += 1 on the first layer).
// One thread per (edge, float4 chunk): coalesced 16B gather + 4 f32 atomics.
// ---------------------------------------------------------------------------
__global__ void sage_scatter(const float* __restrict__ h,
                             const int* __restrict__ src,
                             const int* __restrict__ dst,
                             float* __restrict__ summed,
                             float* __restrict__ cnt,
                             int E, int F, int do_cnt) {
    const int chunks = F >> 2;                       // 96/4 = 24 float4 per edge
    const long long total = (long long)E * chunks;
    const long long tid = (long long)blockIdx.x * blockDim.x + threadIdx.x;
    if (tid >= total) return;
    const int e = (int)(tid / chunks);
    const int c = (int)(tid - (long long)e * chunks);
    const int s = src[e];
    const int d = dst[e];
    const float4 v = *(const float4*)(h + (long long)s * F + c * 4);
    float* o = summed + (long long)d * F + c * 4;
    atomicAdd(o + 0, v.x);
    atomicAdd(o + 1, v.y);
    atomicAdd(o + 2, v.z);
    atomicAdd(o + 3, v.w);
    if (do_cnt && c == 0) atomicAdd(cnt + d, 1.0f);
}

// ---------------------------------------------------------------------------
// Fused SAGE layer GEMM via V_WMMA_F32_16X16X4_F32 (fp32, matches reference):
//   out = act( (summed/max(cnt,1)) @ wl^T  +  h @ wr^T  +  bias )
// One wave32 computes one 16x16 output tile; accumulator seeded with bias.
//
// ISA fragment layouts (CDNA5 ISA):
//   A 16x4  : a[i] = A[m0 + lane&15][k + 2*(lane>>4) + i]
//   B  4x16 : b[i] = B[k + 2*(lane>>4) + i][n0 + lane&15] = W[col][k + ...]
//   C/D     : c[v] -> row m0 + 8*(lane>>4) + v, col n0 + lane&15
// ---------------------------------------------------------------------------
template <bool RELU, int K>
__global__ void sage_gemm(const float* __restrict__ summed,
                          const float* __restrict__ cnt,
                          const float* __restrict__ h,
                          const float* __restrict__ wl,   // [OUT, K] row-major
                          const float* __restrict__ bias, // [OUT]
                          const float* __restrict__ wr,   // [OUT, K] row-major
                          float* __restrict__ out,        // [M, OUT]
                          int mTiles, int OUT) {
    const int lane   = threadIdx.x & 31;
    const int wave   = (int)((blockIdx.x * blockDim.x + threadIdx.x) >> 5);
    const int nWaves = (int)((gridDim.x * blockDim.x) >> 5);
    const int nTiles = OUT >> 4;
    const int total  = mTiles * nTiles;

    const int half = lane >> 4;        // 0 or 1
    const int l16  = lane & 15;
    const int kb   = half * 2;         // K sub-offset for A/B fragments

    for (int t = wave; t < total; t += nWaves) {
        const int mt  = t / nTiles;
        const int nt  = t - mt * nTiles;
        const int row = (mt << 4) + l16;           // A-matrix row for this lane
        const int col = (nt << 4) + l16;           // output column for this lane

        const float inv = 1.0f / fmaxf(cnt[row], 1.0f);

        v8f c;
        const float bv = bias[col];
        #pragma unroll
        for (int i = 0; i < 8; ++i) c[i] = bv;

        const float* sA = summed + (long long)row * K + kb;
        const float* hA = h      + (long long)row * K + kb;
        const float* Bl = wl     + (long long)col * K + kb;
        const float* Br = wr     + (long long)col * K + kb;

        // mean-aggregated neighbors x W_l^T
        #pragma unroll
        for (int k = 0; k < K; k += 4) {
            v2f a, b;
            a[0] = sA[k] * inv;
            a[1] = sA[k + 1] * inv;
            b[0] = Bl[k];
            b[1] = Bl[k + 1];
            c = __builtin_amdgcn_wmma_f32_16x16x4_f32(false, a, false, b,
                                                      (short)0, c, false, false);
        }
        // self features x W_r^T
        #pragma unroll
        for (int k = 0; k < K; k += 4) {
            v2f a, b;
            a[0] = hA[k];
            a[1] = hA[k + 1];
            b[0] = Br[k];
            b[1] = Br[k + 1];
            c = __builtin_amdgcn_wmma_f32_16x16x4_f32(false, a, false, b,
                                                      (short)0, c, false, false);
        }

        const int orow0 = (mt << 4) + half * 8;
        #pragma unroll
        for (int v = 0; v < 8; ++v) {
            float val = c[v];
            if (RELU) val = fmaxf(val, 0.0f);
            out[(long long)(orow0 + v) * OUT + col] = val;
        }
    }
}

// ---------------------------------------------------------------------------
// Host-side launcher
// ---------------------------------------------------------------------------
extern "C" void kernel_launch(void* const* d_in, const int* in_sizes, int n_in,
                              void* d_out, int out_size, void* d_ws, size_t ws_size,
                              hipStream_t stream) {
    const float* x    = (const float*)d_in[0];
    const int*   ei   = (const int*)d_in[1];   // int32 (JAX x64 disabled)
    const float* w1_l = (const float*)d_in[2];
    const float* b1   = (const float*)d_in[3];
    const float* w1_r = (const float*)d_in[4];
    const float* w2_l = (const float*)d_in[5];
    const float* b2   = (const float*)d_in[6];
    const float* w2_r = (const float*)d_in[7];
    float* out = (float*)d_out;

    const int FIN = 96;
    const int N   = in_sizes[0] / FIN;          // 50000
    const int E   = in_sizes[1] / 2;            // 800000
    const int HID = in_sizes[2] / FIN;          // 96
    const int LAT = in_sizes[5] / HID;          // 48

    const int* src = ei;
    const int* dst = ei + E;

    float* ws     = (float*)d_ws;
    float* summed = ws;                              // N*96 (shared by both layers)
    float* cnt    = summed + (size_t)N * FIN;        // N
    float* z      = cnt + N;                         // N*96 (layer-1 output)

    // ---- Layer 1 ----
    hipMemsetAsync(summed, 0, (size_t)N * FIN * sizeof(float), stream);
    hipMemsetAsync(cnt,    0, (size_t)N * sizeof(float), stream);

    {
        const long long work = (long long)E * (FIN / 4);
        const int threads = 256;
        const unsigned blocks = (unsigned)((work + threads - 1) / threads);
        sage_scatter<<<blocks, threads, 0, stream>>>(x, src, dst, summed, cnt,
                                                     E, FIN, 1);
    }

    const int gemmBlocks = 1024, gemmThreads = 256;
    sage_gemm<true, 96><<<gemmBlocks, gemmThreads, 0, stream>>>(
        summed, cnt, x, w1_l, b1, w1_r, z, N / 16, HID);

    // ---- Layer 2 ----
    hipMemsetAsync(summed, 0, (size_t)N * HID * sizeof(float), stream);
    {
        const long long work = (long long)E * (HID / 4);
        const int threads = 256;
        const unsigned blocks = (unsigned)((work + threads - 1) / threads);
        sage_scatter<<<blocks, threads, 0, stream>>>(z, src, dst, summed,
                                                     (float*)nullptr, E, HID, 0);
    }

    sage_gemm<false, 96><<<gemmBlocks, gemmThreads, 0, stream>>>(
        summed, cnt, z, w2_l, b2, w2_r, out, N / 16, LAT);
}